// SAGE_68865505624226
// MI455X (gfx1250) — compile-verified
//
#include <hip/hip_runtime.h>
#include <hip/hip_bf16.h>
#include <math.h>

// Problem constants (match reference setup_inputs)
#define GN 50000    // nodes
#define GE 800000   // edges
#define GF 128      // input features
#define GH 64       // hidden
#define GC 40       // classes

typedef __attribute__((ext_vector_type(2))) float v2f;
typedef __attribute__((ext_vector_type(8))) float v8f;

// D(16x16,f32) = A(16x4,f32) x B(4x16,f32) + C
__device__ __forceinline__ v8f wmma4(v2f a, v2f b, v8f c) {
  return __builtin_amdgcn_wmma_f32_16x16x4_f32(false, a, false, b, (short)0, c,
                                               false, false);
}

// ---------------------------------------------------------------- zero
__global__ __launch_bounds__(256) void SAGE_zero(float* __restrict__ p, int n) {
  int i = blockIdx.x * blockDim.x + threadIdx.x;
  if (i < n) p[i] = 0.0f;
}

// ------------------------------------------------- scatter1: agg1 += x[src]
// one wave32 per edge; each lane moves 4 contiguous floats (full 128-f row)
__global__ __launch_bounds__(256) void SAGE_scatter1(
    const float* __restrict__ x, const long long* __restrict__ ei,
    float* __restrict__ agg1, float* __restrict__ cnt) {
  const int wid = (int)((blockIdx.x * blockDim.x + threadIdx.x) >> 5);
  const int lane = threadIdx.x & 31;
  if (wid >= GE) return;
  const long long src = ei[wid];
  const long long dst = ei[(long long)GE + wid];
  const float4 v = ((const float4*)(x + (size_t)src * GF))[lane];
  float* a = agg1 + (size_t)dst * GF + lane * 4;
  atomicAdd(a + 0, v.x);
  atomicAdd(a + 1, v.y);
  atomicAdd(a + 2, v.z);
  atomicAdd(a + 3, v.w);
  if (lane == 0) atomicAdd(cnt + dst, 1.0f);
}

// ------------------------------------------------- scatter2: agg2 += h[src]
__global__ __launch_bounds__(256) void SAGE_scatter2(
    const float* __restrict__ h, const long long* __restrict__ ei,
    float* __restrict__ agg2) {
  const int wid = (int)((blockIdx.x * blockDim.x + threadIdx.x) >> 5);
  const int lane = threadIdx.x & 31;
  if (wid >= GE) return;
  const long long src = ei[wid];
  const long long dst = ei[(long long)GE + wid];
  const float2 v = ((const float2*)(h + (size_t)src * GH))[lane];
  float* a = agg2 + (size_t)dst * GH + lane * 2;
  atomicAdd(a + 0, v.x);
  atomicAdd(a + 1, v.y);
}

// ---------------------------------------------------------------------------
// Fused conv1 + lin1, one wave32 per 16-node tile.
//   hidden = relu(l2norm(mean @ Wl1 + bl1 + x @ Wr1))     (staged in LDS)
//   h      = relu([x | hidden] @ Wlin + blin)             (written to global)
// ---------------------------------------------------------------------------
__global__ __launch_bounds__(128) void SAGE_conv1_lin1(
    const float* __restrict__ x, const float* __restrict__ agg1,
    const float* __restrict__ cnt, const float* __restrict__ Wl1,
    const float* __restrict__ bl1, const float* __restrict__ Wr1,
    const float* __restrict__ Wlin, const float* __restrict__ blin,
    float* __restrict__ hout, int nTiles) {
  __shared__ float sHid[4][16 * 65];  // per-wave 16x64 tile, padded stride 65

  const int lane = threadIdx.x & 31;
  const int wave = threadIdx.x >> 5;
  const int tile = blockIdx.x * 4 + wave;
  if (tile >= nTiles) return;
  const int base = tile * 16;
  const int m = lane & 15;              // A row / B,D col within 16-group
  const int kh = (lane >> 4) << 1;      // K sub-offset: 0 (lo half) / 2 (hi)
  const int rofs = (lane >> 4) << 3;    // D row offset: 0 / 8
  const int node = base + m;
  const float invc = 1.0f / fmaxf(cnt[node], 1.0f);
  const float* __restrict__ arow = agg1 + (size_t)node * GF;
  const float* __restrict__ xrow = x + (size_t)node * GF;

  v8f acc[4];
#pragma unroll
  for (int t = 0; t < 4; ++t)
#pragma unroll
    for (int r = 0; r < 8; ++r) acc[t][r] = 0.0f;

  // ---- mean @ Wl1 (K = 0..127)
  for (int k0 = 0; k0 < GF; k0 += 4) {
    const int k = k0 + kh;
    v2f a;
    a.x = arow[k] * invc;
    a.y = arow[k + 1] * invc;
#pragma unroll
    for (int t = 0; t < 4; ++t) {
      const int n = t * 16 + m;
      v2f b;
      b.x = Wl1[k * GH + n];
      b.y = Wl1[(k + 1) * GH + n];
      acc[t] = wmma4(a, b, acc[t]);
    }
  }
  // ---- + x @ Wr1
  for (int k0 = 0; k0 < GF; k0 += 4) {
    const int k = k0 + kh;
    v2f a;
    a.x = xrow[k];
    a.y = xrow[k + 1];
#pragma unroll
    for (int t = 0; t < 4; ++t) {
      const int n = t * 16 + m;
      v2f b;
      b.x = Wr1[k * GH + n];
      b.y = Wr1[(k + 1) * GH + n];
      acc[t] = wmma4(a, b, acc[t]);
    }
  }

  // ---- bias + rowwise L2 norm + relu -> LDS staging
  float* hs = &sHid[wave][0];
#pragma unroll
  for (int r = 0; r < 8; ++r) {
    float ss = 0.0f;
#pragma unroll
    for (int t = 0; t < 4; ++t) {
      acc[t][r] += bl1[t * 16 + m];
      ss += acc[t][r] * acc[t][r];
    }
    // reduce across the 16 lanes holding this row (width-16 segments)
    ss += __shfl_xor(ss, 8, 16);
    ss += __shfl_xor(ss, 4, 16);
    ss += __shfl_xor(ss, 2, 16);
    ss += __shfl_xor(ss, 1, 16);
    const float sc = 1.0f / fmaxf(sqrtf(ss), 1e-12f);
    const int row = r + rofs;
#pragma unroll
    for (int t = 0; t < 4; ++t)
      hs[row * 65 + t * 16 + m] = fmaxf(acc[t][r] * sc, 0.0f);
  }

  // ---- lin1: [x | hidden] @ Wlin + blin, relu
  v8f acc2[4];
#pragma unroll
  for (int t = 0; t < 4; ++t)
#pragma unroll
    for (int r = 0; r < 8; ++r) acc2[t][r] = 0.0f;

  for (int k0 = 0; k0 < GF; k0 += 4) {  // x part: Wlin rows 0..127
    const int k = k0 + kh;
    v2f a;
    a.x = xrow[k];
    a.y = xrow[k + 1];
#pragma unroll
    for (int t = 0; t < 4; ++t) {
      const int n = t * 16 + m;
      v2f b;
      b.x = Wlin[k * GH + n];
      b.y = Wlin[(k + 1) * GH + n];
      acc2[t] = wmma4(a, b, acc2[t]);
    }
  }
  for (int k0 = 0; k0 < GH; k0 += 4) {  // hidden part: Wlin rows 128..191
    const int k = k0 + kh;
    v2f a;
    a.x = hs[m * 65 + k];
    a.y = hs[m * 65 + k + 1];
#pragma unroll
    for (int t = 0; t < 4; ++t) {
      const int n = t * 16 + m;
      v2f b;
      b.x = Wlin[(GF + k) * GH + n];
      b.y = Wlin[(GF + k + 1) * GH + n];
      acc2[t] = wmma4(a, b, acc2[t]);
    }
  }

#pragma unroll
  for (int r = 0; r < 8; ++r) {
    const int onode = base + r + rofs;
#pragma unroll
    for (int t = 0; t < 4; ++t) {
      const int n = t * 16 + m;
      hout[(size_t)onode * GH + n] = fmaxf(acc2[t][r] + blin[n], 0.0f);
    }
  }
}

// ---------------------------------------------------------------------------
// conv2: out = l2norm(mean2 @ Wl2 + bl2 + h @ Wr2), cols padded 40 -> 48
// ---------------------------------------------------------------------------
__global__ __launch_bounds__(128) void SAGE_conv2(
    const float* __restrict__ h, const float* __restrict__ agg2,
    const float* __restrict__ cnt, const float* __restrict__ Wl2,
    const float* __restrict__ bl2, const float* __restrict__ Wr2,
    float* __restrict__ out, int nTiles) {
  const int lane = threadIdx.x & 31;
  const int wave = threadIdx.x >> 5;
  const int tile = blockIdx.x * 4 + wave;
  if (tile >= nTiles) return;
  const int base = tile * 16;
  const int m = lane & 15;
  const int kh = (lane >> 4) << 1;
  const int rofs = (lane >> 4) << 3;
  const int node = base + m;
  const float invc = 1.0f / fmaxf(cnt[node], 1.0f);
  const float* __restrict__ arow = agg2 + (size_t)node * GH;
  const float* __restrict__ hrow = h + (size_t)node * GH;

  v8f acc[3];
#pragma unroll
  for (int t = 0; t < 3; ++t)
#pragma unroll
    for (int r = 0; r < 8; ++r) acc[t][r] = 0.0f;

  for (int k0 = 0; k0 < GH; k0 += 4) {  // mean2 @ Wl2
    const int k = k0 + kh;
    v2f a;
    a.x = arow[k] * invc;
    a.y = arow[k + 1] * invc;
#pragma unroll
    for (int t = 0; t < 3; ++t) {
      const int n = t * 16 + m;
      v2f b;
      b.x = (n < GC) ? Wl2[k * GC + n] : 0.0f;
      b.y = (n < GC) ? Wl2[(k + 1) * GC + n] : 0.0f;
      acc[t] = wmma4(a, b, acc[t]);
    }
  }
  for (int k0 = 0; k0 < GH; k0 += 4) {  // + h @ Wr2
    const int k = k0 + kh;
    v2f a;
    a.x = hrow[k];
    a.y = hrow[k + 1];
#pragma unroll
    for (int t = 0; t < 3; ++t) {
      const int n = t * 16 + m;
      v2f b;
      b.x = (n < GC) ? Wr2[k * GC + n] : 0.0f;
      b.y = (n < GC) ? Wr2[(k + 1) * GC + n] : 0.0f;
      acc[t] = wmma4(a, b, acc[t]);
    }
  }

#pragma unroll
  for (int r = 0; r < 8; ++r) {
    float ss = 0.0f;
#pragma unroll
    for (int t = 0; t < 3; ++t) {
      const int n = t * 16 + m;
      if (n < GC) acc[t][r] += bl2[n];  // cols >= 40 stay exactly 0
      ss += acc[t][r] * acc[t][r];
    }
    ss += __shfl_xor(ss, 8, 16);
    ss += __shfl_xor(ss, 4, 16);
    ss += __shfl_xor(ss, 2, 16);
    ss += __shfl_xor(ss, 1, 16);
    const float sc = 1.0f / fmaxf(sqrtf(ss), 1e-12f);
    const int onode = base + r + rofs;
#pragma unroll
    for (int t = 0; t < 3; ++t) {
      const int n = t * 16 + m;
      if (n < GC) out[(size_t)onode * GC + n] = acc[t][r] * sc;
    }
  }
}

// ---------------------------------------------------------------------------
extern "C" void kernel_launch(void* const* d_in, const int* in_sizes, int n_in,
                              void* d_out, int out_size, void* d_ws,
                              size_t ws_size, hipStream_t stream) {
  (void)in_sizes; (void)n_in; (void)out_size; (void)ws_size;
  const float* x      = (const float*)d_in[0];
  const long long* ei = (const long long*)d_in[1];  // int64 edge_index [2,E]
  const float* Wl1    = (const float*)d_in[2];
  const float* bl1    = (const float*)d_in[3];
  const float* Wr1    = (const float*)d_in[4];
  const float* Wlin   = (const float*)d_in[5];
  const float* blin   = (const float*)d_in[6];
  const float* Wl2    = (const float*)d_in[7];
  const float* bl2    = (const float*)d_in[8];
  const float* Wr2    = (const float*)d_in[9];
  float* out = (float*)d_out;

  // workspace layout (floats): agg1[N*F] | cnt[N] | h[N*H] | agg2[N*H]
  float* agg1 = (float*)d_ws;
  float* cnt  = agg1 + (size_t)GN * GF;
  float* hbuf = cnt + GN;
  float* agg2 = hbuf + (size_t)GN * GH;

  const int n1 = GN * GF + GN;  // agg1 + cnt are contiguous
  SAGE_zero<<<(n1 + 255) / 256, 256, 0, stream>>>(agg1, n1);
  const int n2 = GN * GH;
  SAGE_zero<<<(n2 + 255) / 256, 256, 0, stream>>>(agg2, n2);

  const int sblocks = (GE + 7) / 8;  // 8 waves (edges) per 256-thread block
  SAGE_scatter1<<<sblocks, 256, 0, stream>>>(x, ei, agg1, cnt);

  const int nTiles = (GN + 15) / 16;
  const int cblocks = (nTiles + 3) / 4;  // 4 waves (tiles) per 128-thr block
  SAGE_conv1_lin1<<<cblocks, 128, 0, stream>>>(x, agg1, cnt, Wl1, bl1, Wr1,
                                               Wlin, blin, hbuf, nTiles);

  SAGE_scatter2<<<sblocks, 256, 0, stream>>>(hbuf, ei, agg2);

  SAGE_conv2<<<cblocks, 128, 0, stream>>>(hbuf, agg2, cnt, Wl2, bl2, Wr2, out,
                                          nTiles);
}